// DKVMN_53231824667116
// MI455X (gfx1250) — compile-verified
//
#include <hip/hip_runtime.h>
#include <math.h>

// Problem constants (match reference)
#define NUM_Q   1000
#define DIM_K   128
#define DIM_V   128
#define N_SLOTS 50
#define B_SZ    64
#define L_SZ    200
#define M_TOK   (B_SZ * L_SZ)   // 12800 tokens
#define LDS_PAD 4                // row stride 132 floats = 528B: 16B-aligned, bank-stride 4

typedef __attribute__((ext_vector_type(2))) float v2f;
typedef __attribute__((ext_vector_type(8))) float v8f;

// ---------------------------------------------------------------------------
// Kernel 1: gather kt = k_emb[q], vt = v_emb[q + NUM_Q*r]
// ---------------------------------------------------------------------------
__global__ void dkvmn_gather(const int* __restrict__ q, const int* __restrict__ r,
                             const float* __restrict__ k_emb, const float* __restrict__ v_emb,
                             float* __restrict__ kt, float* __restrict__ vt) {
    int tok = blockIdx.x;
    int d   = threadIdx.x;
    int qi  = q[tok];
    int qr  = qi + NUM_Q * r[tok];
    kt[(size_t)tok * DIM_K + d] = k_emb[(size_t)qi * DIM_K + d];
    vt[(size_t)tok * DIM_V + d] = v_emb[(size_t)qr * DIM_V + d];
}

// ---------------------------------------------------------------------------
// Shared helper: cooperative float4 staging of a 16 x KDIM A-tile into LDS.
// Concat mode (A1 != nullptr): A = [A0 | A1], each of row-length KDIM/2.
// ---------------------------------------------------------------------------
template <int KDIM>
__device__ __forceinline__ void stage_A_tile(float (*Alds)[KDIM + LDS_PAD],
                                             const float* __restrict__ A0,
                                             const float* __restrict__ A1,
                                             int m0, int tid) {
    constexpr int V4 = KDIM / 4;               // float4s per row
    for (int idx = tid; idx < 16 * V4; idx += 256) {
        int row = idx / V4;
        int c4  = idx - row * V4;
        int col = c4 * 4;
        float4 v;
        if (A1 != nullptr) {
            const float* src = (col < KDIM / 2)
                ? (A0 + (size_t)(m0 + row) * (KDIM / 2) + col)
                : (A1 + (size_t)(m0 + row) * (KDIM / 2) + (col - KDIM / 2));
            v = *(const float4*)src;
        } else {
            v = *(const float4*)(A0 + (size_t)(m0 + row) * KDIM + col);
        }
        *(float4*)&Alds[row][col] = v;
    }
}

// ---------------------------------------------------------------------------
// Kernel 2: FUSED gate GEMMs with V_WMMA_F32_16X16X4_F32 (native f32):
//   et = sigmoid(vt @ eW + eb),  at = tanh(vt @ aW + ab)    [N = 128]
// One A-fragment load feeds two WMMAs (shared vt tile, two accumulators).
// Block = 256 threads (8 waves); each wave owns one 16-wide n-tile.
// All W offsets are compile-time -> global_load_b32 with immediate offsets.
// ---------------------------------------------------------------------------
__global__ __launch_bounds__(256) void dkvmn_gates(
        const float* __restrict__ vt,
        const float* __restrict__ eW, const float* __restrict__ eb,
        const float* __restrict__ aW, const float* __restrict__ ab,
        float* __restrict__ et, float* __restrict__ at) {
    constexpr int N = 128;
    __shared__ float Alds[16][DIM_V + LDS_PAD];
    const int m0  = blockIdx.x * 16;
    const int tid = threadIdx.x;

    stage_A_tile<DIM_V>(Alds, vt, nullptr, m0, tid);
    __syncthreads();

    const int lane = tid & 31;
    const int n0   = (tid >> 5) * 16;   // wave -> n-tile
    const int nl   = lane & 15;
    const int m    = lane & 15;
    const int kk   = (lane >> 4) * 2;   // lanes 16-31 handle K+2,K+3

    // Per-lane base pointers: loop offsets are compile-time immediates.
    const float* We = eW + kk * N + n0 + nl;
    const float* Wa = aW + kk * N + n0 + nl;

    v8f acc_e = {};
    v8f acc_a = {};
    #pragma unroll
    for (int k0 = 0; k0 < DIM_V; k0 += 4) {
        v2f a;                                   // aligned ds_load_b64
        a.x = Alds[m][k0 + kk];
        a.y = Alds[m][k0 + kk + 1];
        v2f be, ba;
        be.x = We[k0 * N];  be.y = We[(k0 + 1) * N];
        ba.x = Wa[k0 * N];  ba.y = Wa[(k0 + 1) * N];
        acc_e = __builtin_amdgcn_wmma_f32_16x16x4_f32(
                    false, a, false, be, (short)0, acc_e, false, false);
        acc_a = __builtin_amdgcn_wmma_f32_16x16x4_f32(
                    false, a, false, ba, (short)0, acc_a, false, false);
    }

    const float bve = eb[n0 + nl];
    const float bva = ab[n0 + nl];
    const int   mb  = (lane >> 4) * 8;
    #pragma unroll
    for (int j = 0; j < 8; ++j) {
        size_t o = (size_t)(m0 + mb + j) * N + n0 + nl;
        et[o] = 1.0f / (1.0f + __expf(-(acc_e[j] + bve)));
        at[o] = tanhf(acc_a[j] + bva);
    }
}

// ---------------------------------------------------------------------------
// Kernel 3: ft = tanh(concat([rt, kt]) @ fW + fb)   (K = 256, N = 128, WMMA)
// ---------------------------------------------------------------------------
__global__ __launch_bounds__(256) void dkvmn_ft(
        const float* __restrict__ rt, const float* __restrict__ kt,
        const float* __restrict__ fW, const float* __restrict__ fb,
        float* __restrict__ ft) {
    constexpr int KDIM = 2 * DIM_K;
    constexpr int N    = 128;
    __shared__ float Alds[16][KDIM + LDS_PAD];
    const int m0  = blockIdx.x * 16;
    const int tid = threadIdx.x;

    stage_A_tile<KDIM>(Alds, rt, kt, m0, tid);
    __syncthreads();

    const int lane = tid & 31;
    const int n0   = (tid >> 5) * 16;
    const int nl   = lane & 15;
    const int m    = lane & 15;
    const int kk   = (lane >> 4) * 2;

    const float* Wp = fW + kk * N + n0 + nl;

    v8f acc = {};
    #pragma unroll
    for (int k0 = 0; k0 < KDIM; k0 += 4) {
        v2f a;
        a.x = Alds[m][k0 + kk];
        a.y = Alds[m][k0 + kk + 1];
        v2f b;
        b.x = Wp[k0 * N];
        b.y = Wp[(k0 + 1) * N];
        acc = __builtin_amdgcn_wmma_f32_16x16x4_f32(
                  false, a, false, b, (short)0, acc, false, false);
    }

    const float bv = fb[n0 + nl];
    const int   mb = (lane >> 4) * 8;
    #pragma unroll
    for (int j = 0; j < 8; ++j)
        ft[(size_t)(m0 + mb + j) * N + n0 + nl] = tanhf(acc[j] + bv);
}

// ---------------------------------------------------------------------------
// Kernel 4: wt = softmax(kt @ Mk) over N_SLOTS=50.  Tiny -> scalar.
// ---------------------------------------------------------------------------
__global__ void dkvmn_wt_softmax(const float* __restrict__ kt, const float* __restrict__ Mk,
                                 float* __restrict__ wt) {
    __shared__ float sl[N_SLOTS];
    __shared__ float smax, ssum;
    int tok = blockIdx.x;
    int n   = threadIdx.x;
    if (n < N_SLOTS) {
        const float* k = kt + (size_t)tok * DIM_K;
        float s = 0.f;
        #pragma unroll 8
        for (int i = 0; i < DIM_K; ++i) s += k[i] * Mk[i * N_SLOTS + n];
        sl[n] = s;
    }
    __syncthreads();
    if (n == 0) {
        float mx = sl[0];
        for (int i = 1; i < N_SLOTS; ++i) mx = fmaxf(mx, sl[i]);
        smax = mx;
    }
    __syncthreads();
    if (n < N_SLOTS) sl[n] = __expf(sl[n] - smax);
    __syncthreads();
    if (n == 0) {
        float s = 0.f;
        for (int i = 0; i < N_SLOTS; ++i) s += sl[i];
        ssum = s;
    }
    __syncthreads();
    if (n < N_SLOTS) wt[(size_t)tok * N_SLOTS + n] = sl[n] / ssum;
}

// ---------------------------------------------------------------------------
// Kernel 5: sequential memory scan. One thread per (batch b, column v):
// 8192 independent chains, Mv[:,v] (50 floats) in registers.
// Dominant HBM cost: 327.7 MB Mv stores (lane-contiguous over v).
// ---------------------------------------------------------------------------
__global__ __launch_bounds__(DIM_V, 2) void dkvmn_scan(
        const float* __restrict__ wt, const float* __restrict__ et,
        const float* __restrict__ at, const float* __restrict__ Mv0,
        float* __restrict__ rt_out, float* __restrict__ Mv_out) {
    const int b = blockIdx.x;    // 0..63
    const int v = threadIdx.x;   // 0..127
    __shared__ float wsh[N_SLOTS];

    float Mv[N_SLOTS];
    #pragma unroll
    for (int n = 0; n < N_SLOTS; ++n) Mv[n] = Mv0[n * DIM_V + v];

    for (int t = 0; t < L_SZ; ++t) {
        const size_t tok = (size_t)b * L_SZ + t;
        if (v < N_SLOTS) wsh[v] = wt[tok * N_SLOTS + v];
        __syncthreads();

        const float e = et[tok * DIM_V + v];
        const float a = at[tok * DIM_V + v];
        float* out = Mv_out + tok * (size_t)(N_SLOTS * DIM_V) + v;

        float r = 0.f;
        #pragma unroll
        for (int n = 0; n < N_SLOTS; ++n) {
            float w = wsh[n];
            float m = Mv[n];
            r = fmaf(w, m, r);                         // read BEFORE update
            m = m * (1.0f - w * e) + w * a;            // erase + add
            Mv[n] = m;
            out[n * DIM_V] = m;                        // contiguous over v lanes
        }
        rt_out[tok * DIM_V + v] = r;
        __syncthreads();
    }
}

// ---------------------------------------------------------------------------
// Kernel 6: p = sigmoid(ft @ pW + pb). One wave per token, shuffle reduction.
// ---------------------------------------------------------------------------
__global__ void dkvmn_p(const float* __restrict__ ft, const float* __restrict__ pW,
                        const float* __restrict__ pb, float* __restrict__ p) {
    const int wavesPerBlock = blockDim.x >> 5;
    const int tok  = blockIdx.x * wavesPerBlock + (threadIdx.x >> 5);
    const int lane = threadIdx.x & 31;
    float s = 0.f;
    #pragma unroll
    for (int i = lane; i < DIM_K; i += 32)
        s = fmaf(ft[(size_t)tok * DIM_K + i], pW[i], s);
    #pragma unroll
    for (int off = 16; off > 0; off >>= 1) s += __shfl_down(s, off, 32);
    if (lane == 0) p[tok] = 1.0f / (1.0f + __expf(-(s + pb[0])));
}

// ---------------------------------------------------------------------------
// Launch
// ---------------------------------------------------------------------------
extern "C" void kernel_launch(void* const* d_in, const int* in_sizes, int n_in,
                              void* d_out, int out_size, void* d_ws, size_t ws_size,
                              hipStream_t stream) {
    const int*   q     = (const int*)  d_in[0];
    const int*   r     = (const int*)  d_in[1];
    const float* k_emb = (const float*)d_in[2];
    const float* v_emb = (const float*)d_in[3];
    const float* Mk    = (const float*)d_in[4];
    const float* Mv0   = (const float*)d_in[5];
    const float* fW    = (const float*)d_in[6];
    const float* fb    = (const float*)d_in[7];
    const float* pW    = (const float*)d_in[8];
    const float* pb    = (const float*)d_in[9];
    const float* eW    = (const float*)d_in[10];
    const float* eb    = (const float*)d_in[11];
    const float* aW    = (const float*)d_in[12];
    const float* ab    = (const float*)d_in[13];

    float* out    = (float*)d_out;
    float* p_out  = out;                              // [B,L]
    float* Mv_out = out + (size_t)M_TOK;              // [B,L,N_SLOTS,DIM_V]

    float* ws = (float*)d_ws;
    size_t off = 0;
    float* kt = ws + off; off += (size_t)M_TOK * DIM_K;
    float* vt = ws + off; off += (size_t)M_TOK * DIM_V;
    float* et = ws + off; off += (size_t)M_TOK * DIM_V;
    float* at = ws + off; off += (size_t)M_TOK * DIM_V;
    float* wt = ws + off; off += (size_t)M_TOK * N_SLOTS;
    float* rt = ws + off; off += (size_t)M_TOK * DIM_V;
    float* ft = ws + off; off += (size_t)M_TOK * DIM_K;

    // 1. gather embeddings
    dkvmn_gather<<<M_TOK, 128, 0, stream>>>(q, r, k_emb, v_emb, kt, vt);
    // 2. fused gates: et = sigmoid(vt@eW+eb), at = tanh(vt@aW+ab)   (f32 WMMA)
    dkvmn_gates<<<M_TOK / 16, 256, 0, stream>>>(vt, eW, eb, aW, ab, et, at);
    // 3. attention weights: wt = softmax(kt @ Mk)
    dkvmn_wt_softmax<<<M_TOK, 64, 0, stream>>>(kt, Mk, wt);
    // 4. sequential scan (reads gates, writes Mv directly into d_out)
    dkvmn_scan<<<B_SZ, DIM_V, 0, stream>>>(wt, et, at, Mv0, rt, Mv_out);
    // 5. ft = tanh(concat([rt, kt]) @ fW + fb)     (f32 WMMA, K=256 concat)
    dkvmn_ft<<<M_TOK / 16, 256, 0, stream>>>(rt, kt, fW, fb, ft);
    // 6. p = sigmoid(ft @ pW + pb)
    dkvmn_p<<<M_TOK / 8, 256, 0, stream>>>(ft, pW, pb, p_out);
}